// CayleyCirculantSSMLayer_62302795596546
// MI455X (gfx1250) — compile-verified
//
#include <hip/hip_runtime.h>
#include <math.h>

#define D_MODEL 1024
#define STATE   512
#define BATCH   4
#define SEQLEN  8192
#define CHUNK   512
#define NCHUNK  (SEQLEN / CHUNK)   // 16

// GEMM tiling
#define BM 128
#define BN 64
#define BK 32
#define LDSPITCH 36   // padded stride (floats): conflict-free 16-row fragment reads

typedef __attribute__((ext_vector_type(2))) float v2f;
typedef __attribute__((ext_vector_type(8))) float v8f;

// ---------------------------------------------------------------------------
// Kernel 0: Cayley eigenvalues.
// omega_j = Im(FFT(a_full))_j = -sum_k a_full[k] * sin(2*pi*j*k/512)
// lambda_j = (1 - i w)/(1 + i w), |lambda| = 1. Also lambda^CHUNK for carries.
// ---------------------------------------------------------------------------
__global__ void eig_kernel(const float* __restrict__ a, float* __restrict__ eig) {
    int j = threadIdx.x;                       // 512 threads, one per channel
    float w = 0.f;
    const float twopi_n = 6.28318530717958647692f / (float)STATE;
    for (int k = 1; k < STATE; ++k) {
        float af = (k <= STATE / 2) ? a[k - 1] : -a[STATE - k];
        int ph = (j * k) & (STATE - 1);        // exact periodic reduction
        w -= af * sinf(twopi_n * (float)ph);
    }
    float den = 1.f + w * w;
    float c = (1.f - w * w) / den;             // Re(lambda)
    float s = -2.f * w / den;                  // Im(lambda)
    float th = atan2f(s, c);
    eig[j * 4 + 0] = c;
    eig[j * 4 + 1] = s;
    eig[j * 4 + 2] = cosf((float)CHUNK * th);  // Re(lambda^CHUNK)
    eig[j * 4 + 3] = sinf((float)CHUNK * th);  // Im(lambda^CHUNK)
}

// ---------------------------------------------------------------------------
// FP32 WMMA GEMM with LDS double buffering:
//   C[m,n] = sum_k A[m,k] * W[n,k]  (+ optional D[n]*U[m,n])
// A: M x K row-major, W: N x K row-major.
// V_WMMA_F32_16X16X4_F32 fragment layouts (ISA 7.12.2):
//   A 16x4: lane L -> row M=L%16, VGPR0/1 = K = 2*(L/16)+{0,1}
//   B 4x16: lane L -> col N=L%16, VGPR0/1 = K = 2*(L/16)+{0,1}
// Fragments are read from LDS as 8-byte (v2f) loads, bank-conflict-free with
// the 36-float pitch. Global loads are coalesced float4 row segments.
// ---------------------------------------------------------------------------
__global__ __launch_bounds__(256)
void wmma_gemm_nt(const float* __restrict__ A, const float* __restrict__ W,
                  float* __restrict__ C, int M, int N, int K,
                  const float* __restrict__ skipU,   // M x N or nullptr
                  const float* __restrict__ skipD) { // N     or nullptr
    __shared__ float As[2][BM * LDSPITCH];   // 2 x 18 KB
    __shared__ float Ws[2][BN * LDSPITCH];   // 2 x  9 KB

    const int tid  = threadIdx.x;
    const int lane = tid & 31;
    const int wave = tid >> 5;
    const int m0 = blockIdx.x * BM;
    const int n0 = blockIdx.y * BN;
    const int r16 = lane & 15;                 // fragment row (A) / col (B)
    const int kb  = (lane >> 4) << 1;          // K sub-offset: 0 or 2

    // staging assignments: float4 chunks of 32-float row segments
    // A panel: 128 rows x 8 chunks = 1024 chunks -> 4 per thread
    // W panel:  64 rows x 8 chunks =  512 chunks -> 2 per thread
    float4 ra[4], rw[2];

#define FETCH(k0)                                                              \
    {                                                                          \
        _Pragma("unroll")                                                      \
        for (int i = 0; i < 4; ++i) {                                          \
            int ch = tid + i * 256, row = ch >> 3, c4 = (ch & 7) << 2;         \
            ra[i] = *(const float4*)(A + (size_t)(m0 + row) * K + (k0) + c4);  \
        }                                                                      \
        _Pragma("unroll")                                                      \
        for (int i = 0; i < 2; ++i) {                                          \
            int ch = tid + i * 256, row = ch >> 3, c4 = (ch & 7) << 2;         \
            rw[i] = *(const float4*)(W + (size_t)(n0 + row) * K + (k0) + c4);  \
        }                                                                      \
    }

#define STAGE(buf)                                                             \
    {                                                                          \
        _Pragma("unroll")                                                      \
        for (int i = 0; i < 4; ++i) {                                          \
            int ch = tid + i * 256, row = ch >> 3, c4 = (ch & 7) << 2;         \
            float* p = &As[buf][row * LDSPITCH + c4];                          \
            *(v2f*)p       = (v2f){ra[i].x, ra[i].y};                          \
            *(v2f*)(p + 2) = (v2f){ra[i].z, ra[i].w};                          \
        }                                                                      \
        _Pragma("unroll")                                                      \
        for (int i = 0; i < 2; ++i) {                                          \
            int ch = tid + i * 256, row = ch >> 3, c4 = (ch & 7) << 2;         \
            float* p = &Ws[buf][row * LDSPITCH + c4];                          \
            *(v2f*)p       = (v2f){rw[i].x, rw[i].y};                          \
            *(v2f*)(p + 2) = (v2f){rw[i].z, rw[i].w};                          \
        }                                                                      \
    }

    v8f acc[4] = {};

    FETCH(0);
    STAGE(0);
    __syncthreads();

    const int P = K / BK;
    for (int p = 0; p < P; ++p) {
        const int buf = p & 1;
        if (p + 1 < P) FETCH((p + 1) * BK);    // overlap HBM latency w/ WMMAs

        const float* Ab = &As[buf][(wave * 16 + r16) * LDSPITCH + kb];
        const float* Wb = &Ws[buf][r16 * LDSPITCH + kb];
#pragma unroll
        for (int kk = 0; kk < BK; kk += 4) {
            v2f a = *(const v2f*)(Ab + kk);
#pragma unroll
            for (int t = 0; t < 4; ++t) {
                v2f b = *(const v2f*)(Wb + t * 16 * LDSPITCH + kk);
                acc[t] = __builtin_amdgcn_wmma_f32_16x16x4_f32(
                    false, a, false, b, (short)0, acc[t], false, false);
            }
        }

        if (p + 1 < P) STAGE(buf ^ 1);
        __syncthreads();
    }

    // C/D layout: VGPR r -> row m0+wave*16 + r + 8*(lane/16), col n0+16t+lane%16
    const int rowHalf = (lane >> 4) * 8;
    const int mBase = m0 + wave * 16 + rowHalf;
#pragma unroll
    for (int t = 0; t < 4; ++t) {
        int col = n0 + t * 16 + r16;
        float dsc = skipD ? skipD[col] : 0.f;
#pragma unroll
        for (int r = 0; r < 8; ++r) {
            size_t idx = (size_t)(mBase + r) * N + col;
            float v = acc[t][r];
            if (skipU) v += dsc * skipU[idx];
            C[idx] = v;
        }
    }
#undef FETCH
#undef STAGE
}

// ---------------------------------------------------------------------------
// Chunked parallel scan of h[t] = lambda*h[t-1] + Bu[t]  (complex, |lambda|=1)
// Thread index = ((b*NCHUNK + c) * STATE) + j, j fastest -> coalesced.
// ---------------------------------------------------------------------------
__global__ void scan_carry_kernel(const float* __restrict__ Bu,
                                  const float* __restrict__ eig,
                                  float* __restrict__ Lc) {
    int idx = blockIdx.x * blockDim.x + threadIdx.x;   // BATCH*NCHUNK*STATE
    int j = idx & (STATE - 1);
    int c = (idx >> 9) & (NCHUNK - 1);
    int b = idx >> 13;
    float lc = eig[j * 4 + 0], ls = eig[j * 4 + 1];
    const float* x = Bu + ((size_t)b * SEQLEN + (size_t)c * CHUNK) * STATE + j;
    float hr = 0.f, hi = 0.f;
    for (int t = 0; t < CHUNK; ++t) {
        float xv = x[(size_t)t * STATE];
        float nr = lc * hr - ls * hi + xv;
        hi = ls * hr + lc * hi;
        hr = nr;
    }
    Lc[(size_t)idx * 2 + 0] = hr;
    Lc[(size_t)idx * 2 + 1] = hi;
}

__global__ void scan_combine_kernel(const float* __restrict__ Lc,
                                    const float* __restrict__ eig,
                                    float* __restrict__ Gin) {
    int idx = blockIdx.x * blockDim.x + threadIdx.x;   // BATCH*STATE
    int j = idx & (STATE - 1);
    int b = idx >> 9;
    float pc = eig[j * 4 + 2], ps = eig[j * 4 + 3];    // lambda^CHUNK
    float gr = 0.f, gi = 0.f;
    for (int c = 0; c < NCHUNK; ++c) {
        size_t o = ((size_t)b * NCHUNK + c) * STATE + j;
        Gin[o * 2 + 0] = gr;                            // carry-in for chunk c
        Gin[o * 2 + 1] = gi;
        float lr = Lc[o * 2 + 0], li = Lc[o * 2 + 1];
        float nr = lr + pc * gr - ps * gi;
        gi = li + ps * gr + pc * gi;
        gr = nr;
    }
}

__global__ void scan_apply_kernel(const float* __restrict__ Bu,
                                  const float* __restrict__ eig,
                                  const float* __restrict__ Gin,
                                  float* __restrict__ Hre) {
    int idx = blockIdx.x * blockDim.x + threadIdx.x;
    int j = idx & (STATE - 1);
    int c = (idx >> 9) & (NCHUNK - 1);
    int b = idx >> 13;
    float lc = eig[j * 4 + 0], ls = eig[j * 4 + 1];
    size_t base = ((size_t)b * SEQLEN + (size_t)c * CHUNK) * STATE + j;
    const float* x = Bu + base;
    float* y = Hre + base;
    float hr = Gin[(size_t)idx * 2 + 0];
    float hi = Gin[(size_t)idx * 2 + 1];
    for (int t = 0; t < CHUNK; ++t) {
        float xv = x[(size_t)t * STATE];
        float nr = lc * hr - ls * hi + xv;
        hi = ls * hr + lc * hi;
        hr = nr;
        y[(size_t)t * STATE] = hr;                      // only Re(h) is needed
    }
}

// ---------------------------------------------------------------------------
extern "C" void kernel_launch(void* const* d_in, const int* in_sizes, int n_in,
                              void* d_out, int out_size, void* d_ws, size_t ws_size,
                              hipStream_t stream) {
    const float* u        = (const float*)d_in[0];  // (4, 8192, 1024)
    const float* a_params = (const float*)d_in[1];  // (256,)
    const float* B_w      = (const float*)d_in[2];  // (512, 1024)
    const float* C_w      = (const float*)d_in[3];  // (1024, 512)
    const float* Dv       = (const float*)d_in[4];  // (1024,)
    float* y  = (float*)d_out;                      // (4, 8192, 1024)
    float* ws = (float*)d_ws;

    const size_t BTN = (size_t)BATCH * SEQLEN * STATE;   // 16.78M floats
    float* Bu  = ws;                                     // B x T x N
    float* Hre = Bu + BTN;                               // B x T x N
    float* eig = Hre + BTN;                              // 512 x 4
    float* Lc  = eig + STATE * 4;                        // B x NCHUNK x N x 2
    float* Gin = Lc + (size_t)BATCH * NCHUNK * STATE * 2;

    // 1) per-channel eigenvalues
    eig_kernel<<<1, STATE, 0, stream>>>(a_params, eig);

    // 2) Bu = u @ B_w^T    (M=32768, N=512, K=1024)
    dim3 g1((BATCH * SEQLEN) / BM, STATE / BN);
    wmma_gemm_nt<<<g1, 256, 0, stream>>>(u, B_w, Bu,
                                         BATCH * SEQLEN, STATE, D_MODEL,
                                         nullptr, nullptr);

    // 3) chunked complex scan -> Hre = Re(h)
    int nscan = BATCH * NCHUNK * STATE;                  // 32768 chains
    scan_carry_kernel<<<nscan / 256, 256, 0, stream>>>(Bu, eig, Lc);
    scan_combine_kernel<<<(BATCH * STATE) / 256, 256, 0, stream>>>(Lc, eig, Gin);
    scan_apply_kernel<<<nscan / 256, 256, 0, stream>>>(Bu, eig, Gin, Hre);

    // 4) y = Hre @ C_w^T + D*u   (M=32768, N=1024, K=512)
    dim3 g2((BATCH * SEQLEN) / BM, D_MODEL / BN);
    wmma_gemm_nt<<<g2, 256, 0, stream>>>(Hre, C_w, y,
                                         BATCH * SEQLEN, D_MODEL, STATE,
                                         u, Dv);
}